// Proposal_Target_layer_11450382811681
// MI455X (gfx1250) — compile-verified
//
#include <hip/hip_runtime.h>
#include <stdint.h>

// ---------------- constants (match reference) ----------------
#define N_SAMPLE 512
#define N_FG     128          // 512 * 0.25
#define N_BG     384
#define HBINS    4096
#define CAP      4096         // candidate buffer per class (sorted in LDS)
#define FG_THRESH 0.5f
#define BG_LOW    0.1f
#define BG_HIGH   0.5f
#define MAX_GT    512
#define RPT      4            // ROIs per thread in the IoU kernel

__device__ __forceinline__ int score_bin(float s) {
    int b = (int)(s * (float)HBINS);
    if (b > HBINS - 1) b = HBINS - 1;
    if (b < 0) b = 0;
    return b;
}

// ---------------- kernel 1: zero control structures ----------------
__global__ void ptl_init(int* __restrict__ hist_fg, int* __restrict__ hist_bg,
                         int* __restrict__ ctrl) {
    int i = blockIdx.x * blockDim.x + threadIdx.x;
    if (i < HBINS) { hist_fg[i] = 0; hist_bg[i] = 0; }
    if (i < 16) ctrl[i] = 0;
}

// ---------------- kernel 2: IoU row-max/argmax + score histograms ----------------
// GT table (8 KB) staged into LDS with CDNA5 async global->LDS loads.
// Register-blocked: each thread carries RPT=4 ROIs so every LDS broadcast of a
// GT box is amortized over 4 IoU evaluations (~56 VALU : 2 DS per iteration).
__global__ __launch_bounds__(256)
void ptl_iou(const float4* __restrict__ rois, const float4* __restrict__ gt,
             int N, int G,
             float* __restrict__ max_row, int* __restrict__ argmax_row,
             int* __restrict__ hist_fg, int* __restrict__ hist_bg) {
    __shared__ float4 sgt[MAX_GT];
    __shared__ float  sarea[MAX_GT];
    const int t = threadIdx.x;

    // CDNA5 async path: each lane DMAs one 16B gt box straight into LDS.
    for (int g = t; g < G; g += blockDim.x) {
        uint64_t ga = (uint64_t)(uintptr_t)(gt + g);
        uint32_t la = (uint32_t)(uintptr_t)(&sgt[g]);   // low 32 bits == LDS offset
        asm volatile("global_load_async_to_lds_b128 %0, %1, off"
                     :: "v"(la), "v"(ga) : "memory");
    }
    asm volatile("s_wait_asynccnt 0" ::: "memory");
    __syncthreads();

    for (int g = t; g < G; g += blockDim.x) {
        float4 b = sgt[g];
        sarea[g] = (b.z - b.x) * (b.w - b.y);
    }
    __syncthreads();

    const int base = blockIdx.x * (256 * RPT);

    int    idx[RPT];
    float4 a[RPT];
    float  areaA[RPT];
    float  best[RPT];
    int    bidx[RPT];

#pragma unroll
    for (int r = 0; r < RPT; ++r) {
        int i = base + r * 256 + t;
        idx[r] = i;
        int ic = (i < N) ? i : (N - 1);          // clamp loads for the tail
        a[r] = rois[ic];
        areaA[r] = (a[r].z - a[r].x) * (a[r].w - a[r].y);
        best[r] = -1.0f;
        bidx[r] = 0;
    }

#pragma unroll 2
    for (int g = 0; g < G; ++g) {
        float4 b  = sgt[g];
        float  ab = sarea[g];
#pragma unroll
        for (int r = 0; r < RPT; ++r) {
            float ltx = fmaxf(a[r].x, b.x);
            float lty = fmaxf(a[r].y, b.y);
            float rbx = fminf(a[r].z, b.z);
            float rby = fminf(a[r].w, b.w);
            float w = fmaxf(rbx - ltx, 0.0f);
            float h = fmaxf(rby - lty, 0.0f);
            float inter = w * h;
            float uni   = areaA[r] + ab - inter;
            float iou   = inter * __builtin_amdgcn_rcpf(uni); // v_rcp_f32
            if (iou > best[r]) { best[r] = iou; bidx[r] = g; } // first-max tie rule
        }
    }

#pragma unroll
    for (int r = 0; r < RPT; ++r) {
        int i = idx[r];
        if (i < N) {
            max_row[i]    = best[r];
            argmax_row[i] = bidx[r];
            if (best[r] > FG_THRESH)
                atomicAdd(&hist_fg[score_bin(best[r])], 1);
            else if (best[r] > BG_LOW && best[r] < BG_HIGH)
                atomicAdd(&hist_bg[score_bin(best[r])], 1);
        }
    }
}

// ---------------- kernel 3: find k-th-value bin thresholds ----------------
// ctrl layout: [0]=cnt_fg [1]=cnt_bg [2]=T_fg [3]=T_bg
__global__ void ptl_thresh(const int* __restrict__ hist_fg,
                           const int* __restrict__ hist_bg,
                           int* __restrict__ ctrl) {
    if (threadIdx.x == 0) {
        int acc = 0, T = 0;
        for (int b = HBINS - 1; b >= 0; --b) {
            acc += hist_fg[b];
            if (acc >= N_FG) { T = b; break; }
        }
        ctrl[2] = T;
    }
    if (threadIdx.x == 1) {
        int acc = 0, T = 0;
        for (int b = HBINS - 1; b >= 0; --b) {
            acc += hist_bg[b];
            if (acc >= N_BG) { T = b; break; }
        }
        ctrl[3] = T;
    }
}

// ---------------- kernel 4: compact candidates >= threshold bin ----------------
// 64-bit key = (score_bits << 32) | ~idx  ->  descending sort == score desc, idx asc.
__global__ __launch_bounds__(256)
void ptl_compact(const float* __restrict__ max_row, int N,
                 int* __restrict__ ctrl,
                 unsigned long long* __restrict__ cand_fg,
                 unsigned long long* __restrict__ cand_bg) {
    int i = blockIdx.x * blockDim.x + threadIdx.x;
    if (i >= N) return;
    float s = max_row[i];
    if (s > FG_THRESH) {
        if (score_bin(s) >= ctrl[2]) {
            int p = atomicAdd(&ctrl[0], 1);
            if (p < CAP)
                cand_fg[p] = ((unsigned long long)__float_as_uint(s) << 32)
                           | (unsigned)(0xFFFFFFFFu - (unsigned)i);
        }
    } else if (s > BG_LOW && s < BG_HIGH) {
        if (score_bin(s) >= ctrl[3]) {
            int p = atomicAdd(&ctrl[1], 1);
            if (p < CAP)
                cand_bg[p] = ((unsigned long long)__float_as_uint(s) << 32)
                           | (unsigned)(0xFFFFFFFFu - (unsigned)i);
        }
    }
}

// ---------------- kernel 5: LDS bitonic sort, pick top-k for fg then bg ----------------
__global__ __launch_bounds__(1024)
void ptl_select(const unsigned long long* __restrict__ cand_fg,
                const unsigned long long* __restrict__ cand_bg,
                const int* __restrict__ ctrl,
                int* __restrict__ sel) {
    __shared__ unsigned long long keys[CAP];    // 32 KB
    const int tid = threadIdx.x;

    for (int phase = 0; phase < 2; ++phase) {
        const unsigned long long* cand = phase ? cand_bg : cand_fg;
        int cnt = ctrl[phase];
        if (cnt > CAP) cnt = CAP;
        const int k = phase ? N_BG : N_FG;

        // fill: pads get score 0 with ascending fake indices (mimics top_k over -1 ties)
        for (int s = tid; s < CAP; s += blockDim.x)
            keys[s] = (s < cnt) ? cand[s]
                                : (unsigned long long)(0xFFFFFFFFu - (unsigned)(s - cnt));
        __syncthreads();

        // bitonic sort, descending
        for (int kk = 2; kk <= CAP; kk <<= 1) {
            for (int j = kk >> 1; j > 0; j >>= 1) {
                for (int s = tid; s < CAP; s += blockDim.x) {
                    int ixj = s ^ j;
                    if (ixj > s) {
                        unsigned long long A = keys[s], B = keys[ixj];
                        bool desc = ((s & kk) == 0);
                        if (desc ? (A < B) : (A > B)) { keys[s] = B; keys[ixj] = A; }
                    }
                }
                __syncthreads();
            }
        }

        for (int s = tid; s < k; s += blockDim.x) {
            unsigned idx = 0xFFFFFFFFu - (unsigned)(keys[s] & 0xFFFFFFFFull);
            sel[(phase ? N_FG : 0) + s] = (int)idx;
        }
        __syncthreads();   // protect keys[] before next phase refill
    }
}

// ---------------- kernel 6: gather + bbox transform + normalize + labels ----------------
__global__ __launch_bounds__(512)
void ptl_final(const float4* __restrict__ rois, const float4* __restrict__ gt,
               const int* __restrict__ gt_labels,
               const float* __restrict__ max_row, const int* __restrict__ argmax_row,
               const int* __restrict__ sel,
               float* __restrict__ out_delta, int* __restrict__ out_labels) {
    int s = blockIdx.x * blockDim.x + threadIdx.x;
    if (s >= N_SAMPLE) return;
    int idx = sel[s];
    int m   = argmax_row[idx];
    float4 p = rois[idx];
    float4 g = gt[m];

    float pw  = p.z - p.x, ph  = p.w - p.y;
    float pcx = p.x + 0.5f * pw, pcy = p.y + 0.5f * ph;
    float gw  = g.z - g.x, gh  = g.w - g.y;
    float gcx = g.x + 0.5f * gw, gcy = g.y + 0.5f * gh;

    float dx = (gcx - pcx) / pw;
    float dy = (gcy - pcy) / ph;
    float dw = __logf(gw / pw);
    float dh = __logf(gh / ph);

    // (delta - MEAN)/STD with MEAN=0, STD=[0.1,0.1,0.2,0.2]
    out_delta[s * 4 + 0] = dx * 10.0f;
    out_delta[s * 4 + 1] = dy * 10.0f;
    out_delta[s * 4 + 2] = dw * 5.0f;
    out_delta[s * 4 + 3] = dh * 5.0f;

    bool valid_fg = (s < N_FG) && (max_row[idx] > FG_THRESH);
    out_labels[s] = valid_fg ? gt_labels[m] : 0;
}

// ---------------- launcher ----------------
extern "C" void kernel_launch(void* const* d_in, const int* in_sizes, int n_in,
                              void* d_out, int out_size, void* d_ws, size_t ws_size,
                              hipStream_t stream) {
    const float* rois      = (const float*)d_in[0];
    const float* gt        = (const float*)d_in[1];
    const int*   gt_labels = (const int*)d_in[2];   // jax int64 falls back to int32 (x64 off)

    const int N = in_sizes[0] / 4;   // 200000
    const int G = in_sizes[1] / 4;   // 512 (<= MAX_GT)

    // workspace carve-up (~1.7 MB)
    char* w = (char*)d_ws;
    float* max_row = (float*)w;                 w += (size_t)N * sizeof(float);
    int*   argmax  = (int*)w;                   w += (size_t)N * sizeof(int);
    int*   hist_fg = (int*)w;                   w += HBINS * sizeof(int);
    int*   hist_bg = (int*)w;                   w += HBINS * sizeof(int);
    int*   ctrl    = (int*)w;                   w += 16 * sizeof(int);
    unsigned long long* cand_fg = (unsigned long long*)w; w += CAP * sizeof(unsigned long long);
    unsigned long long* cand_bg = (unsigned long long*)w; w += CAP * sizeof(unsigned long long);
    int*   sel     = (int*)w;                   w += N_SAMPLE * sizeof(int);

    float* out_delta  = (float*)d_out;                       // [512,4]
    int*   out_labels = ((int*)d_out) + N_SAMPLE * 4;        // [512] int32 bits

    const int roisPerBlock = 256 * RPT;

    ptl_init   <<<(HBINS + 255) / 256, 256, 0, stream>>>(hist_fg, hist_bg, ctrl);
    ptl_iou    <<<(N + roisPerBlock - 1) / roisPerBlock, 256, 0, stream>>>(
                   (const float4*)rois, (const float4*)gt, N, G,
                   max_row, argmax, hist_fg, hist_bg);
    ptl_thresh <<<1, 64, 0, stream>>>(hist_fg, hist_bg, ctrl);
    ptl_compact<<<(N + 255) / 256, 256, 0, stream>>>(max_row, N, ctrl, cand_fg, cand_bg);
    ptl_select <<<1, 1024, 0, stream>>>(cand_fg, cand_bg, ctrl, sel);
    ptl_final  <<<(N_SAMPLE + 511) / 512, 512, 0, stream>>>(
                   (const float4*)rois, (const float4*)gt, gt_labels,
                   max_row, argmax, sel, out_delta, out_labels);
}